// FullOnlineTDCLLayer_11708080849227
// MI455X (gfx1250) — compile-verified
//
#include <hip/hip_runtime.h>

typedef float v2f __attribute__((ext_vector_type(2)));
typedef float v8f __attribute__((ext_vector_type(8)));

constexpr int B = 64, K = 4096, L = 4096;
constexpr float BETA = 0.9f, GAMMA = 0.1f, TEMP = 0.5f, INPUT_THR = 0.02f, LR = 0.01f;
constexpr float PI_F = 3.14159265358979323846f;
constexpr int KT = K / 16;   // 256 k-tiles
// ---- d_out layout (flat concat in reference return order, all f32) ----
constexpr size_t O_SPK   = 0;                       // (B,K)
constexpr size_t O_LOSS  = O_SPK + (size_t)B * K;   // scalar
constexpr size_t O_SIM   = O_LOSS + 1;              // (B,)
constexpr size_t O_TRACE = O_SIM + B;               // (B,K)
constexpr size_t O_USP   = O_TRACE + (size_t)B * K; // scalar
constexpr size_t O_TSP   = O_USP + 1;               // scalar
constexpr size_t O_W     = O_TSP + 1;               // (K,L)
constexpr size_t O_INTR  = O_W + (size_t)K * L;     // (B,L)
// ---- workspace layout (floats) ----
constexpr size_t WS_SURR = 0;                        // (B,K)
constexpr size_t WS_AT   = WS_SURR + (size_t)B * K;  // (K,B)  a^T
constexpr size_t WS_SIMP = WS_AT + (size_t)K * B;    // (B,KT) sim partials
constexpr size_t WS_DL   = WS_SIMP + (size_t)B * KT; // (B,)
constexpr size_t WS_ZCNT = WS_DL + B;                // 1 int

__device__ __forceinline__ v8f wmma_f32(v2f a, v2f b, v8f c) {
  // V_WMMA_F32_16X16X4_F32: D(16x16,f32) = A(16x4,f32) * B(4x16,f32) + C
  return __builtin_amdgcn_wmma_f32_16x16x4_f32(false, a, false, b, (short)0, c,
                                               false, false);
}

__global__ void k0_zero(float* ws) {
  *(int*)(ws + WS_ZCNT) = 0;
}

// GEMM1: mem_out = beta*mem_prev + x @ W^T, fused spk/surr/sim-partials.
// grid (4, 128), block 128 (4 waves). Each block: one b-tile, two k-tiles;
// the L=4096 contraction is split 4 ways across waves, reduced through LDS.
__global__ void k1_gemm1(const float* __restrict__ x, const float* __restrict__ W,
                         const float* __restrict__ mem_prev,
                         const float* __restrict__ trace,
                         float* __restrict__ out, float* __restrict__ ws) {
  const int tid  = threadIdx.x;
  const int lane = tid & 31;
  const int wave = tid >> 5;        // 0..3 -> L-quarter
  const int n  = lane & 15;         // column within tile / A row
  const int hi = lane >> 4;         // half-wave select (K-slice {0,1} vs {2,3})
  const int b0 = blockIdx.x * 16;
  const int kt = blockIdx.y * 2;    // first of two k-tiles
  const int k0 = kt * 16;

  const int lbeg = wave * (L / 4);
  const int lend = lbeg + (L / 4);

  // A: x[b0+n][l + {0,1} or {2,3}]   B: W[k0(+16)+n][l + {0,1} or {2,3}]
  const float* xrow  = x + (size_t)(b0 + n) * L + 2 * hi;
  const float* wrow0 = W + (size_t)(k0 + n) * L + 2 * hi;
  const float* wrow1 = wrow0 + (size_t)16 * L;

  v8f acc0 = {0.f, 0.f, 0.f, 0.f, 0.f, 0.f, 0.f, 0.f};
  v8f acc1 = acc0;
#pragma unroll 4
  for (int l = lbeg; l < lend; l += 4) {
    v2f a   = *(const v2f*)(xrow + l);
    v2f bb0 = *(const v2f*)(wrow0 + l);
    v2f bb1 = *(const v2f*)(wrow1 + l);
    acc0 = wmma_f32(a, bb0, acc0);
    acc1 = wmma_f32(a, bb1, acc1);
  }

  // Cross-wave reduction of the partial accumulators (8 KB LDS).
  __shared__ float red[4][16][32];
#pragma unroll
  for (int j = 0; j < 8; ++j) {
    red[wave][j][lane]     = acc0[j];
    red[wave][8 + j][lane] = acc1[j];
  }
  __syncthreads();

  if (wave == 0) {
    int zc = 0;
#pragma unroll
    for (int t = 0; t < 2; ++t) {
      const int kcol = k0 + t * 16 + n;
#pragma unroll
      for (int j = 0; j < 8; ++j) {
        float accv = red[0][t * 8 + j][lane] + red[1][t * 8 + j][lane] +
                     red[2][t * 8 + j][lane] + red[3][t * 8 + j][lane];
        const int m = j + 8 * hi;                    // C/D: M = j (+8 for hi)
        const size_t idx = (size_t)(b0 + m) * K + kcol;
        float mem = BETA * mem_prev[idx] + accv;
        float spk = mem > 1.0f ? 1.0f : 0.0f;
        float d = PI_F * (mem - 1.0f);
        float surr = 1.0f / (1.0f + d * d);
        out[O_SPK + idx] = spk;
        ws[WS_SURR + idx] = surr;
        float tr = trace[idx];
        zc += (tr == 0.0f) ? 1 : 0;
        float s = spk * tr;                          // reduce over 16 cols
        s += __shfl_xor(s, 1, 32);
        s += __shfl_xor(s, 2, 32);
        s += __shfl_xor(s, 4, 32);
        s += __shfl_xor(s, 8, 32);
        if (n == 0) ws[WS_SIMP + (size_t)(b0 + m) * KT + (kt + t)] = s;
      }
    }
    for (int off = 1; off < 32; off <<= 1) zc += __shfl_xor(zc, off, 32);
    if (lane == 0) atomicAdd((int*)(ws + WS_ZCNT), zc);
  }
}

// Per-batch scalars: sim_score, loss, dL, loss_mean, sparsities. 1 block x 64.
__global__ void k2_scalars(const float* __restrict__ input_sum,
                           float* __restrict__ out, float* __restrict__ ws) {
  __shared__ float sl[B];
  __shared__ float sz[B];
  const int b = threadIdx.x;
  float sum = 0.0f;
  for (int t = 0; t < KT; ++t) sum += ws[WS_SIMP + (size_t)b * KT + t];
  float sim = sum * (1.0f / ((float)K * TEMP));   // mean over K, /TEMP
  out[O_SIM + b] = sim;
  float is = input_sum[b];
  bool act = is > INPUT_THR;
  float loss = fmaxf(GAMMA * is - sim, 0.0f);
  loss = act ? loss : 0.0f;
  ws[WS_DL + b] = (act && loss > 0.0f) ? 1.0f : 0.0f;
  sl[b] = loss;
  sz[b] = (loss == 0.0f) ? 1.0f : 0.0f;
  __syncthreads();
  for (int s = B / 2; s > 0; s >>= 1) {
    if (b < s) { sl[b] += sl[b + s]; sz[b] += sz[b + s]; }
    __syncthreads();
  }
  if (b == 0) {
    out[O_LOSS] = sl[0] * (1.0f / (float)B);
    out[O_USP]  = sz[0] * (1.0f / (float)B);
    int zcnt = *(const int*)(ws + WS_ZCNT);
    out[O_TSP] = (float)zcnt / (float)((size_t)B * K);
  }
}

// a^T[k,b] = dL[b]*trace[b,k]*surr[b,k] via LDS-tiled transpose (coalesced
// both sides); also copies trace to the output. grid K/64, block 256.
__global__ void k3a_build_at(const float* __restrict__ trace,
                             float* __restrict__ out, float* __restrict__ ws) {
  __shared__ float tile[64][65];    // pad: conflict-free column reads
  const int tid = threadIdx.x;
  const int k0 = blockIdx.x * 64;
  // read phase: consecutive tids sweep k (coalesced)
  const int kc = tid & 63;
  const int bq = tid >> 6;          // 0..3
#pragma unroll
  for (int r = 0; r < 16; ++r) {
    const int b = bq * 16 + r;
    const size_t idx = (size_t)b * K + k0 + kc;
    float tr = trace[idx];
    out[O_TRACE + idx] = tr;
    tile[b][kc] = ws[WS_DL + b] * tr * ws[WS_SURR + idx];
  }
  __syncthreads();
  // write phase: consecutive tids sweep b (coalesced)
  const int bc = tid & 63;
  const int kq = tid >> 6;
#pragma unroll
  for (int r = 0; r < 16; ++r) {
    const int k = kq * 16 + r;
    ws[WS_AT + (size_t)(k0 + k) * B + bc] = tile[bc][k];
  }
}

// in_trace = beta*in_trace_prev + x
__global__ void k3b_intrace(const float* __restrict__ x,
                            const float* __restrict__ in_trace_prev,
                            float* __restrict__ out) {
  const size_t idx = (size_t)blockIdx.x * blockDim.x + threadIdx.x;
  out[O_INTR + idx] = BETA * in_trace_prev[idx] + x[idx];
}

// GEMM2 + update: W_new = W + (LR/(TEMP*B*K)) * (a^T @ in_trace).
// grid (256, 128), block 32. Each wave: one k-tile, two l-tiles.
__global__ void k4_gemm2(const float* __restrict__ W, float* __restrict__ out,
                         const float* __restrict__ ws) {
  const int lane = threadIdx.x & 31;
  const int n = lane & 15;
  const int hi = lane >> 4;
  const int k0 = blockIdx.x * 16;
  const int l0 = blockIdx.y * 32;

  const float* aT   = ws + WS_AT;
  const float* intr = out + O_INTR;   // written by k3b earlier on this stream

  // A: aT[k0+n][b + {0,1} or {2,3}]  (contiguous float2 along B)
  const float* arow = aT + (size_t)(k0 + n) * B + 2 * hi;

  v8f acc0 = {0.f, 0.f, 0.f, 0.f, 0.f, 0.f, 0.f, 0.f};
  v8f acc1 = acc0;
#pragma unroll
  for (int b = 0; b < B; b += 4) {
    v2f a = *(const v2f*)(arow + b);
    const int bc = b + 2 * hi;         // B rows for this half-wave
    v2f bb0, bb1;
    bb0.x = intr[(size_t)bc * L + l0 + n];
    bb0.y = intr[(size_t)(bc + 1) * L + l0 + n];
    bb1.x = intr[(size_t)bc * L + l0 + 16 + n];
    bb1.y = intr[(size_t)(bc + 1) * L + l0 + 16 + n];
    acc0 = wmma_f32(a, bb0, acc0);
    acc1 = wmma_f32(a, bb1, acc1);
  }

  // dW = -acc/(B*K); W_new = W - (LR/TEMP)*dW = W + C1*acc
  constexpr float C1 = LR / (TEMP * (float)B * (float)K);
#pragma unroll
  for (int j = 0; j < 8; ++j) {
    const int k = k0 + j + 8 * hi;
    const size_t i0 = (size_t)k * L + l0 + n;
    out[O_W + i0]      = W[i0]      + C1 * acc0[j];
    out[O_W + i0 + 16] = W[i0 + 16] + C1 * acc1[j];
  }
}

extern "C" void kernel_launch(void* const* d_in, const int* in_sizes, int n_in,
                              void* d_out, int out_size, void* d_ws, size_t ws_size,
                              hipStream_t stream) {
  (void)in_sizes; (void)n_in; (void)out_size; (void)ws_size;
  const float* x             = (const float*)d_in[0];
  const float* W             = (const float*)d_in[1];
  const float* mem_prev      = (const float*)d_in[2];
  const float* in_trace_prev = (const float*)d_in[3];
  const float* trace         = (const float*)d_in[4];
  const float* input_sum     = (const float*)d_in[5];
  float* out = (float*)d_out;
  float* ws  = (float*)d_ws;

  k0_zero<<<1, 1, 0, stream>>>(ws);
  k1_gemm1<<<dim3(B / 16, KT / 2), 128, 0, stream>>>(x, W, mem_prev, trace, out, ws);
  k2_scalars<<<1, B, 0, stream>>>(input_sum, out, ws);
  k3a_build_at<<<K / 64, 256, 0, stream>>>(trace, out, ws);
  k3b_intrace<<<(B * L) / 256, 256, 0, stream>>>(x, in_trace_prev, out);
  k4_gemm2<<<dim3(K / 16, L / 32), 32, 0, stream>>>(W, out, ws);
}